// CIP_44392781971895
// MI455X (gfx1250) — compile-verified
//
#include <hip/hip_runtime.h>
#include <hip/hip_bf16.h>

typedef __attribute__((ext_vector_type(2))) float v2f;
typedef __attribute__((ext_vector_type(8))) float v8f;

#define B_ 32
#define S_ 16
#define N_ 32
#define T_ 2000
#define M_ 2
#define SN_ 512            // S*N
#define K1_ 1024           // 2*SN (z and z^2 halves)
#define ROWS_ 64           // B*M
#define COLS2_ 160         // S*Y
#define KC1_ 4             // split-K chunks for GEMM1 (1024/4 = 256)
#define KL1_ 256
#define KC2_ 10            // split-K chunks for GEMM2 (2000/10 = 200)
#define KL2_ 200

// ---------------------------------------------------------------------------
// Kernel 1: build A[64 x 1024] = [ z | z^2 ], z = mean + eps * exp(0.5*log_var)
// row = b*M + m (matches reference transpose to (B,M,S,N))
// ---------------------------------------------------------------------------
__global__ void k_build_A(const float* __restrict__ mean,
                          const float* __restrict__ log_var,
                          const float* __restrict__ eps,
                          float* __restrict__ A) {
    int i = blockIdx.x * blockDim.x + threadIdx.x;   // 0 .. 64*512-1
    if (i >= ROWS_ * SN_) return;
    int row = i >> 9;          // bm
    int sn  = i & (SN_ - 1);
    int b = row >> 1;
    int m = row & 1;
    float mu = mean[b * SN_ + sn];
    float lv = log_var[b * SN_ + sn];
    float e  = eps[(m * B_ + b) * SN_ + sn];
    float z  = mu + e * __expf(0.5f * lv);
    A[row * K1_ + sn]        = z;
    A[row * K1_ + SN_ + sn]  = z * z;
}

// ---------------------------------------------------------------------------
// Kernel 2: build W[2000 x 1024] = [ 2*mu*w | -w ], w = 1/(2*var),
// and bias[t] = sum_sn ( log(4.13273) + 0.5*lv0[n] - 0.5*log(2*pi*var) - mu^2*w )
// ---------------------------------------------------------------------------
__global__ void k_build_W(const float* __restrict__ mean_T,
                          const float* __restrict__ log_var_T,
                          float* __restrict__ W,
                          float* __restrict__ bias) {
    __shared__ float red[256];
    int t   = blockIdx.x;
    int tid = threadIdx.x;
    const float LOG_A   = 1.4189379f;   // ln(4.13273)
    const float LOG_2PI = 1.8378771f;   // ln(2*pi)
    float partial = 0.0f;
    #pragma unroll
    for (int it = 0; it < 2; ++it) {
        int sn  = tid + it * 256;
        float lv  = log_var_T[t * SN_ + sn];
        float var = fmaxf(__expf(lv), 1e-20f);
        float w   = 0.5f / var;
        float mu  = mean_T[t * SN_ + sn];
        W[t * K1_ + sn]       = 2.0f * mu * w;
        W[t * K1_ + SN_ + sn] = -w;
        float lv0 = log_var_T[sn & (N_ - 1)];          // log_var_T[0,0,n]
        float c   = LOG_A + 0.5f * lv0 - 0.5f * (LOG_2PI + __logf(var));
        partial += c - mu * mu * w;
    }
    red[tid] = partial;
    __syncthreads();
    for (int s = 128; s > 0; s >>= 1) {
        if (tid < s) red[tid] += red[tid + s];
        __syncthreads();
    }
    if (tid == 0) bias[t] = red[0];
}

// ---------------------------------------------------------------------------
// Kernel 3: GEMM1 (split-K) via V_WMMA_F32_16X16X4_F32.
// 2000 waves: 125 n-tiles x 4 m-tiles x 4 K-chunks of 256.
// Each wave writes an 8-VGPR partial tile to P1[kc][row][t].
// ---------------------------------------------------------------------------
__global__ void k_gemm1(const float* __restrict__ A,
                        const float* __restrict__ W,
                        float* __restrict__ P1) {
    int wid  = (blockIdx.x * blockDim.x + threadIdx.x) >> 5;   // 0..1999
    int lane = threadIdx.x & 31;
    int tm  = wid & 3;          // 0..3
    int tmp = wid >> 2;         // 0..499
    int kc  = tmp & 3;          // 0..3
    int tn  = tmp >> 2;         // 0..124
    int l16  = lane & 15;
    int hi   = lane >> 4;       // 0 or 1
    int koff = hi << 1;         // 0 or 2
    int mrow = tm * 16 + l16;
    int ncol = tn * 16 + l16;
    const float* Ap = A + mrow * K1_ + kc * KL1_ + koff;
    const float* Wp = W + ncol * K1_ + kc * KL1_ + koff;
    v8f acc = {};
    #pragma unroll 4
    for (int k = 0; k < KL1_; k += 4) {
        v2f av = *(const v2f*)(Ap + k);
        v2f bv = *(const v2f*)(Wp + k);
        acc = __builtin_amdgcn_wmma_f32_16x16x4_f32(
            false, av, false, bv, (short)0, acc, false, false);
    }
    int rbase = tm * 16 + (hi << 3);
    float* Pp = P1 + (size_t)kc * (ROWS_ * T_);
    #pragma unroll
    for (int v = 0; v < 8; ++v)
        Pp[(rbase + v) * T_ + ncol] = acc[v];
}

// ---------------------------------------------------------------------------
// Kernel 4: combine split-K partials: J = min(exp(sum_kc P1 + bias_t), 1e20)
// ---------------------------------------------------------------------------
__global__ void k_combine1(const float* __restrict__ P1,
                           const float* __restrict__ bias,
                           float* __restrict__ J) {
    int i = blockIdx.x * blockDim.x + threadIdx.x;   // 0 .. 64*2000-1
    if (i >= ROWS_ * T_) return;
    int t = i % T_;
    float L = P1[i] + P1[i + ROWS_ * T_] + P1[i + 2 * ROWS_ * T_]
            + P1[i + 3 * ROWS_ * T_] + bias[t];
    J[i] = fminf(__expf(L), 1e20f);
}

// ---------------------------------------------------------------------------
// Kernel 5: den[row] = sum_t J[row][t]   (one wave per row)
// ---------------------------------------------------------------------------
__global__ void k_rowsum(const float* __restrict__ J, float* __restrict__ den) {
    int wave = (blockIdx.x * blockDim.x + threadIdx.x) >> 5;   // 0..63
    int lane = threadIdx.x & 31;
    float s = 0.0f;
    for (int t = lane; t < T_; t += 32) s += J[wave * T_ + t];
    for (int off = 16; off > 0; off >>= 1) s += __shfl_down(s, off, 32);
    if (lane == 0) den[wave] = s;
}

// ---------------------------------------------------------------------------
// Kernel 6: GEMM2 (split-K) via WMMA: num = J[64x2000] * Y[2000x160]
// 400 waves: 10 n-tiles x 4 m-tiles x 10 K-chunks of 200.
// ---------------------------------------------------------------------------
__global__ void k_gemm2(const float* __restrict__ J,
                        const float* __restrict__ Y,
                        float* __restrict__ P2) {
    int wid  = (blockIdx.x * blockDim.x + threadIdx.x) >> 5;   // 0..399
    int lane = threadIdx.x & 31;
    int tm  = wid & 3;          // 0..3
    int tmp = wid >> 2;         // 0..99
    int tn  = tmp % 10;         // 0..9
    int kc  = tmp / 10;         // 0..9
    int l16  = lane & 15;
    int hi   = lane >> 4;
    int koff = hi << 1;
    int mrow = tm * 16 + l16;
    int ncol = tn * 16 + l16;
    int kbase = kc * KL2_;
    const float* Jp = J + mrow * T_ + kbase + koff;
    acc_loop: ;
    v8f acc = {};
    #pragma unroll 4
    for (int k = 0; k < KL2_; k += 4) {
        v2f av = *(const v2f*)(Jp + k);
        v2f bv;
        bv.x = Y[(kbase + k + koff)     * COLS2_ + ncol];
        bv.y = Y[(kbase + k + koff + 1) * COLS2_ + ncol];
        acc = __builtin_amdgcn_wmma_f32_16x16x4_f32(
            false, av, false, bv, (short)0, acc, false, false);
    }
    int rbase = tm * 16 + (hi << 3);
    float* Pp = P2 + (size_t)kc * (ROWS_ * COLS2_);
    #pragma unroll
    for (int v = 0; v < 8; ++v)
        Pp[(rbase + v) * COLS2_ + ncol] = acc[v];
}

// ---------------------------------------------------------------------------
// Kernel 7: combine GEMM2 partials into num[64 x 160]
// ---------------------------------------------------------------------------
__global__ void k_combine2(const float* __restrict__ P2,
                           float* __restrict__ num) {
    int i = blockIdx.x * blockDim.x + threadIdx.x;   // 0 .. 10239
    if (i >= ROWS_ * COLS2_) return;
    float s = 0.0f;
    #pragma unroll
    for (int kc = 0; kc < KC2_; ++kc)
        s += P2[i + kc * (ROWS_ * COLS2_)];
    num[i] = s;
}

// ---------------------------------------------------------------------------
// Kernel 8: out[b][s][y] = clip( mean_m( max(num,1e-20)/max(den,1e-20) ), 0, 1)
// ---------------------------------------------------------------------------
__global__ void k_final(const float* __restrict__ num,
                        const float* __restrict__ den,
                        float* __restrict__ out) {
    int i = blockIdx.x * blockDim.x + threadIdx.x;   // 0..5119
    if (i >= B_ * COLS2_) return;
    int b   = i / COLS2_;
    int col = i - b * COLS2_;
    int r0 = b * 2, r1 = r0 + 1;
    float p0 = fmaxf(num[r0 * COLS2_ + col], 1e-20f) / fmaxf(den[r0], 1e-20f);
    float p1 = fmaxf(num[r1 * COLS2_ + col], 1e-20f) / fmaxf(den[r1], 1e-20f);
    float p  = 0.5f * (p0 + p1);
    out[i] = fminf(fmaxf(p, 0.0f), 1.0f);
}

extern "C" void kernel_launch(void* const* d_in, const int* in_sizes, int n_in,
                              void* d_out, int out_size, void* d_ws, size_t ws_size,
                              hipStream_t stream) {
    (void)in_sizes; (void)n_in; (void)out_size;
    const float* mean      = (const float*)d_in[0];
    const float* log_var   = (const float*)d_in[1];
    const float* mean_T    = (const float*)d_in[2];
    const float* log_var_T = (const float*)d_in[3];
    const float* y_true_T  = (const float*)d_in[4];
    const float* eps       = (const float*)d_in[5];
    float* out = (float*)d_out;

    float* ws   = (float*)d_ws;
    float* A    = ws;                            // 64*1024          =    65,536
    float* W    = A    + ROWS_ * K1_;            // 2000*1024        = 2,048,000
    float* bias = W    + T_ * K1_;               // 2000
    float* P1   = bias + T_;                     // 4*64*2000        =   512,000
    float* J    = P1   + KC1_ * ROWS_ * T_;      // 64*2000          =   128,000
    float* P2   = J    + ROWS_ * T_;             // 10*64*160        =   102,400
    float* num  = P2   + KC2_ * ROWS_ * COLS2_;  // 64*160           =    10,240
    float* den  = num  + ROWS_ * COLS2_;         // 64
    size_t need = (size_t)(den + ROWS_ - ws) * sizeof(float);
    if (ws_size < need) return;

    k_build_A<<<(ROWS_ * SN_ + 255) / 256, 256, 0, stream>>>(mean, log_var, eps, A);
    k_build_W<<<T_, 256, 0, stream>>>(mean_T, log_var_T, W, bias);
    k_gemm1<<<500, 128, 0, stream>>>(A, W, P1);             // 2000 waves
    k_combine1<<<(ROWS_ * T_ + 255) / 256, 256, 0, stream>>>(P1, bias, J);
    k_rowsum<<<8, 256, 0, stream>>>(J, den);                // 64 waves
    k_gemm2<<<100, 128, 0, stream>>>(J, y_true_T, P2);      // 400 waves
    k_combine2<<<(ROWS_ * COLS2_ + 255) / 256, 256, 0, stream>>>(P2, num);
    k_final<<<(B_ * COLS2_ + 255) / 256, 256, 0, stream>>>(num, den, out);
}